// MySDGNN_77738908058039
// MI455X (gfx1250) — compile-verified
//
#include <hip/hip_runtime.h>
#include <math.h>

typedef float v2f __attribute__((ext_vector_type(2)));
typedef float v8f __attribute__((ext_vector_type(8)));

#define D 64
#define NEG_SLOPE 0.2f

// ---- float <-> order-preserving uint encoding for atomic float max ----
__device__ __forceinline__ unsigned f2ord(float f) {
    unsigned u = __float_as_uint(f);
    return (u & 0x80000000u) ? ~u : (u | 0x80000000u);
}
__device__ __forceinline__ float ord2f(unsigned u) {
    return (u & 0x80000000u) ? __uint_as_float(u & 0x7FFFFFFFu)
                             : __uint_as_float(~u);
}

// ---- combined[:,0:64] = x ; combined[:,64+r*64+c] = bg[l][r][c] ----
__global__ void k_init_combined(const float* __restrict__ x,
                                const float* __restrict__ bg_l,   // [R*64]
                                float* __restrict__ comb, int N, int C) {
    size_t gid = (size_t)blockIdx.x * blockDim.x + threadIdx.x;
    size_t total = (size_t)N * C;
    if (gid >= total) return;
    int col = (int)(gid % C);
    size_t n = gid / C;
    float v = (col < D) ? x[n * D + col] : bg_l[col - D];
    comb[gid] = v;
}

__global__ void k_init_md(unsigned* __restrict__ mord, float* __restrict__ denom, int N) {
    int i = blockIdx.x * blockDim.x + threadIdx.x;
    if (i < N) { mord[i] = f2ord(-__builtin_inff()); denom[i] = 0.f; }
}

// ---- h = x @ W (N x 64 = N x 64 * 64 x 64), fused alpha_s/alpha_d ----
// Block: 128 threads = 4 waves; each wave computes a 16x16 tile; block covers 16 rows x 64 cols.
__global__ void __launch_bounds__(128)
k_gemm_h_alpha(const float* __restrict__ x, const float* __restrict__ W,
               const float* __restrict__ asrc, const float* __restrict__ adst,
               float* __restrict__ h, float* __restrict__ alpha_s,
               float* __restrict__ alpha_d, int N) {
    __shared__ float xs[16][D];
    __shared__ float hs[16][D + 1];
    const int grow = blockIdx.x * 16;
    const int tid  = threadIdx.x;

    for (int i = tid; i < 16 * D; i += 128) {
        int r = i / D, c = i % D;
        int row = grow + r;
        xs[r][c] = (row < N) ? x[(size_t)row * D + c] : 0.f;
    }
    __syncthreads();

    const int wave = tid >> 5;
    const int lane = tid & 31;
    const int half = lane >> 4;   // lanes 0-15 hold K=k,k+1 ; lanes 16-31 hold K=k+2,k+3
    const int l15  = lane & 15;
    const int colbase = wave * 16;

    v8f acc = {};
    for (int k = 0; k < D; k += 4) {
        int kk = k + half * 2;
        v2f a, b;
        a.x = xs[l15][kk];
        a.y = xs[l15][kk + 1];
        b.x = W[(size_t)kk * D + colbase + l15];
        b.y = W[(size_t)(kk + 1) * D + colbase + l15];
        acc = __builtin_amdgcn_wmma_f32_16x16x4_f32(false, a, false, b,
                                                    (short)0, acc, false, false);
    }

#pragma unroll
    for (int j = 0; j < 8; ++j) {
        int row = half * 8 + j;            // C layout: lanes 0-15 -> M=j ; 16-31 -> M=8+j
        int col = colbase + l15;
        hs[row][col] = acc[j];
        int g = grow + row;
        if (g < N) h[(size_t)g * D + col] = acc[j];
    }
    __syncthreads();

    if (tid < 16) {
        int g = grow + tid;
        if (g < N) {
            float s1 = 0.f, s2 = 0.f;
            for (int c = 0; c < D; ++c) {
                float v = hs[tid][c];
                s1 += v * asrc[c];
                s2 += v * adst[c];
            }
            alpha_s[g] = s1;
            alpha_d[g] = s2;
        }
    }
}

// ---- pass 1: e = leaky_relu(as[s]+ad[d]); segment max over dst ----
__global__ void k_edge_max(const long long* __restrict__ src,
                           const long long* __restrict__ dst,
                           const float* __restrict__ as, const float* __restrict__ ad,
                           unsigned* __restrict__ mord, float* __restrict__ ebuf,
                           long long E, long long Etot) {
    long long gid = (long long)blockIdx.x * blockDim.x + threadIdx.x;
    if (gid >= Etot) return;
    long long s, d;
    if (gid < E) { s = src[gid]; d = dst[gid]; }
    else         { s = d = gid - E; }         // self loop
    float v = as[s] + ad[d];
    v = (v > 0.f) ? v : NEG_SLOPE * v;
    ebuf[gid] = v;
    atomicMax(&mord[d], f2ord(v));
}

// ---- pass 2: ex = exp(e - m[d]); segment sum -> denom ----
__global__ void k_edge_expsum(const long long* __restrict__ dst,
                              const unsigned* __restrict__ mord,
                              float* __restrict__ ebuf, float* __restrict__ denom,
                              long long E, long long Etot) {
    long long gid = (long long)blockIdx.x * blockDim.x + threadIdx.x;
    if (gid >= Etot) return;
    long long d = (gid < E) ? dst[gid] : (gid - E);
    float ex = __expf(ebuf[gid] - ord2f(mord[d]));
    ebuf[gid] = ex;
    atomicAdd(&denom[d], ex);
}

// ---- pass 3: out[d,:] += h[s,:] * (ex/denom[d]); one wave per edge ----
__global__ void k_edge_scatter(const long long* __restrict__ src,
                               const long long* __restrict__ dst,
                               const float* __restrict__ ebuf,
                               const float* __restrict__ denom,
                               const float* __restrict__ h,
                               float* __restrict__ comb, int slot, int C,
                               long long E, long long Etot) {
    long long gid  = (long long)blockIdx.x * blockDim.x + threadIdx.x;
    long long eidx = gid >> 5;
    int lane = (int)(gid & 31);
    if (eidx >= Etot) return;
    long long s, d;
    if (eidx < E) { s = src[eidx]; d = dst[eidx]; }
    else          { s = d = eidx - E; }
    float w = ebuf[eidx] / denom[d];
    float* outp = comb + (size_t)d * C + D + (size_t)slot * D;
    const float* hp = h + (size_t)s * D;
    atomicAdd(&outp[lane],      hp[lane]      * w);
    atomicAdd(&outp[lane + 32], hp[lane + 32] * w);
}

// ---- x_out = tanh(combined @ W1 + b1) @ W2 + b2 ----
__global__ void __launch_bounds__(128)
k_mlp(const float* __restrict__ comb, const float* __restrict__ W1,
      const float* __restrict__ b1, const float* __restrict__ W2,
      const float* __restrict__ b2, float* __restrict__ out, int N, int C) {
    extern __shared__ float smem[];            // cs[16][C] then hid[16][D+1]
    float* cs  = smem;
    float* hid = smem + 16 * C;
    const int grow = blockIdx.x * 16;
    const int tid  = threadIdx.x;

    for (int i = tid; i < 16 * C; i += 128) {
        int r = i / C, c = i % C;
        int row = grow + r;
        cs[r * C + c] = (row < N) ? comb[(size_t)row * C + c] : 0.f;
    }
    __syncthreads();

    const int wave = tid >> 5;
    const int lane = tid & 31;
    const int half = lane >> 4;
    const int l15  = lane & 15;
    const int colbase = wave * 16;

    v8f acc = {};
    for (int k = 0; k < C; k += 4) {
        int kk = k + half * 2;
        v2f a, b;
        a.x = cs[l15 * C + kk];
        a.y = cs[l15 * C + kk + 1];
        b.x = W1[(size_t)kk * D + colbase + l15];
        b.y = W1[(size_t)(kk + 1) * D + colbase + l15];
        acc = __builtin_amdgcn_wmma_f32_16x16x4_f32(false, a, false, b,
                                                    (short)0, acc, false, false);
    }
#pragma unroll
    for (int j = 0; j < 8; ++j) {
        int row = half * 8 + j;
        int col = colbase + l15;
        hid[row * (D + 1) + col] = tanhf(acc[j] + b1[col]);
    }
    __syncthreads();

    v8f acc2 = {};
    for (int k = 0; k < D; k += 4) {
        int kk = k + half * 2;
        v2f a, b;
        a.x = hid[l15 * (D + 1) + kk];
        a.y = hid[l15 * (D + 1) + kk + 1];
        b.x = W2[(size_t)kk * D + colbase + l15];
        b.y = W2[(size_t)(kk + 1) * D + colbase + l15];
        acc2 = __builtin_amdgcn_wmma_f32_16x16x4_f32(false, a, false, b,
                                                     (short)0, acc2, false, false);
    }
#pragma unroll
    for (int j = 0; j < 8; ++j) {
        int row = half * 8 + j;
        int col = colbase + l15;
        int g = grow + row;
        if (g < N) out[(size_t)g * D + col] = acc2[j] + b2[col];
    }
}

extern "C" void kernel_launch(void* const* d_in, const int* in_sizes, int n_in,
                              void* d_out, int out_size, void* d_ws, size_t ws_size,
                              hipStream_t stream) {
    const float*     x0    = (const float*)d_in[0];
    const long long* edges = (const long long*)d_in[1];   // [R,2,E] int64
    const float*     Wg    = (const float*)d_in[2];       // [L,R,D,D]
    const float*     a_src = (const float*)d_in[3];       // [L,R,D]
    const float*     a_dst = (const float*)d_in[4];       // [L,R,D]
    const float*     bg    = (const float*)d_in[5];       // [L,R,D]
    const float*     W1    = (const float*)d_in[6];       // [L,5D,D]
    const float*     b1    = (const float*)d_in[7];       // [L,D]
    const float*     W2    = (const float*)d_in[8];       // [L,D,D]
    const float*     b2    = (const float*)d_in[9];       // [L,D]

    const int  N  = in_sizes[0] / D;
    const int  L  = in_sizes[6] / (5 * D * D);
    const int  R  = in_sizes[2] / (L * D * D);
    const long long E    = (long long)in_sizes[1] / ((long long)R * 2);
    const long long Etot = E + N;
    const int  C  = (1 + R) * D;   // combined width (320)

    // workspace carve-up (256B aligned slabs)
    char*  base = (char*)d_ws;
    size_t off  = 0;
    auto take = [&](size_t bytes) -> char* {
        char* p = base + off;
        off = (off + bytes + 255) & ~(size_t)255;
        return p;
    };
    float*    comb  = (float*)   take((size_t)N * C * sizeof(float));
    float*    h     = (float*)   take((size_t)N * D * sizeof(float));
    float*    as    = (float*)   take((size_t)N * sizeof(float));
    float*    ad    = (float*)   take((size_t)N * sizeof(float));
    unsigned* mord  = (unsigned*)take((size_t)N * sizeof(unsigned));
    float*    denom = (float*)   take((size_t)N * sizeof(float));
    float*    ebuf  = (float*)   take((size_t)Etot * sizeof(float));
    float*    xbuf  = (float*)   take((size_t)N * D * sizeof(float));
    (void)ws_size;

    const int TB = 256;
    const int gemm_blocks = (N + 15) / 16;
    const size_t mlp_lds  = (size_t)(16 * C + 16 * (D + 1)) * sizeof(float);

    for (int l = 0; l < L; ++l) {
        const float* x_cur = (l == 0) ? x0 : xbuf;
        float*       x_out = (l == L - 1) ? (float*)d_out : xbuf;

        {
            size_t total = (size_t)N * C;
            k_init_combined<<<(unsigned)((total + TB - 1) / TB), TB, 0, stream>>>(
                x_cur, bg + (size_t)l * R * D, comb, N, C);
        }

        for (int r = 0; r < R; ++r) {
            const float* Wlr  = Wg    + ((size_t)(l * R + r)) * D * D;
            const float* asrc = a_src + ((size_t)(l * R + r)) * D;
            const float* adst = a_dst + ((size_t)(l * R + r)) * D;
            const long long* srcp = edges + ((size_t)r * 2 + 0) * E;
            const long long* dstp = edges + ((size_t)r * 2 + 1) * E;

            k_init_md<<<(N + TB - 1) / TB, TB, 0, stream>>>(mord, denom, N);

            k_gemm_h_alpha<<<gemm_blocks, 128, 0, stream>>>(
                x_cur, Wlr, asrc, adst, h, as, ad, N);

            k_edge_max<<<(unsigned)((Etot + TB - 1) / TB), TB, 0, stream>>>(
                srcp, dstp, as, ad, mord, ebuf, E, Etot);

            k_edge_expsum<<<(unsigned)((Etot + TB - 1) / TB), TB, 0, stream>>>(
                dstp, mord, ebuf, denom, E, Etot);

            long long sthreads = Etot * 32;
            k_edge_scatter<<<(unsigned)((sthreads + TB - 1) / TB), TB, 0, stream>>>(
                srcp, dstp, ebuf, denom, h, comb, r, C, E, Etot);
        }

        k_mlp<<<gemm_blocks, 128, mlp_lds, stream>>>(
            comb, W1 + (size_t)l * 5 * D * D, b1 + (size_t)l * D,
            W2 + (size_t)l * D * D, b2 + (size_t)l * D, x_out, N, C);
    }
}